// NormalizingFlowModel_77180562309404
// MI455X (gfx1250) — compile-verified
//
#include <hip/hip_runtime.h>
#include <hip/hip_bf16.h>

#define NN 1024
#define KK 1024
#define DD 16
#define D2C 8
#define HH 32
#define NLAYERS 4

typedef __attribute__((ext_vector_type(2))) float v2f;
typedef __attribute__((ext_vector_type(4))) float v4f;
typedef __attribute__((ext_vector_type(8))) float v8f;

// ---------------------------------------------------------------------------
// CDNA5 WMMA: D(16x16 f32) = A(16x4 f32) * B(4x16 f32) + C
// ---------------------------------------------------------------------------
__device__ __forceinline__ v8f wmma4(v2f a, v2f b, v8f c) {
  return __builtin_amdgcn_wmma_f32_16x16x4_f32(
      /*neg_a=*/false, a, /*neg_b=*/false, b,
      /*c_mod=*/(short)0, c, /*reuse_a=*/false, /*reuse_b=*/false);
}

// Cross-row (lane ^ 16) exchange. Prefer v_permlanex16_b32 (pure VALU, no LDS
// pipe) over ds_bpermute; identity selectors make row0 lane i read row1 lane i.
__device__ __forceinline__ float xor16(float v) {
#if __has_builtin(__builtin_amdgcn_permlanex16)
  int i = __float_as_int(v);
  int r = __builtin_amdgcn_permlanex16(i, i, 0x76543210, 0xfedcba98,
                                       /*fi=*/true, /*bound_ctrl=*/false);
  return __int_as_float(r);
#else
  return __shfl_xor(v, 16, 32);
#endif
}

// Extract B-operand chunk k (matrix rows 4k..4k+3) from a D-layout 16x16 tile
// held as d[0..7] (lanes 0-15: rows 0-7 at col=lane; lanes 16-31: rows 8-15).
// B layout: lane<16 holds B[4k+0..1][col], lane>=16 holds B[4k+2..3][col].
__device__ __forceinline__ v2f bchunk(const float d[8], int k, bool hi) {
  int g = (k & 1) * 4;
  float p0 = d[g + 0], p1 = d[g + 1], q0 = d[g + 2], q1 = d[g + 3];
  v2f b;
  if (k < 2) {                       // rows 0..7 region
    float s0 = xor16(q0), s1 = xor16(q1);
    b.x = hi ? s0 : p0;
    b.y = hi ? s1 : p1;
  } else {                           // rows 8..15 region
    float s0 = xor16(p0), s1 = xor16(p1);
    b.x = hi ? q0 : s0;
    b.y = hi ? q1 : s1;
  }
  return b;
}

// Fast transcendental paths.
#define LOG2E 1.4426950408889634f
__device__ __forceinline__ float fexp(float x) {
  return __builtin_amdgcn_exp2f(x * LOG2E);
}
__device__ __forceinline__ float ftanh(float x) {
#if __has_builtin(__builtin_amdgcn_tanhf)
  return __builtin_amdgcn_tanhf(x);        // native V_TANH_F32 (1 TRANS op)
#elif __has_builtin(__builtin_amdgcn_tanh_f32)
  return __builtin_amdgcn_tanh_f32(x);
#else
  float e = __builtin_amdgcn_exp2f(x * (2.0f * LOG2E));
  return 1.0f - 2.0f * __builtin_amdgcn_rcpf(1.0f + e);
#endif
}

// ---------------------------------------------------------------------------
// Kernel 1: per-n Cholesky of cov[:,:,n], inverse of L, logdet -> workspace
// cov layout: (D, D, N) -> cov[(i*D+j)*N + n]  (coalesced over n)
// ---------------------------------------------------------------------------
__global__ __launch_bounds__(256) void nf_chol(const float* __restrict__ cov,
                                               float* __restrict__ linv_ws,
                                               float* __restrict__ logdet_ws) {
  int n = blockIdx.x * blockDim.x + threadIdx.x;
  if (n >= NN) return;

  float L[DD][DD];
  for (int i = 0; i < DD; ++i)
    for (int j = 0; j < DD; ++j)
      L[i][j] = cov[((size_t)(i * DD + j)) * NN + n];

  float logdet = 0.0f;
  for (int j = 0; j < DD; ++j) {
    float s = L[j][j];
    for (int k = 0; k < j; ++k) s -= L[j][k] * L[j][k];
    float dj = __builtin_sqrtf(s);
    L[j][j] = dj;
    logdet += 2.0f * __logf(dj);
    float inv = 1.0f / dj;
    for (int i = j + 1; i < DD; ++i) {
      float s2 = L[i][j];
      for (int k = 0; k < j; ++k) s2 -= L[i][k] * L[j][k];
      L[i][j] = s2 * inv;
    }
  }

  float Li[DD][DD];
  for (int i = 0; i < DD; ++i)
    for (int j = 0; j < DD; ++j) Li[i][j] = 0.0f;
  for (int j = 0; j < DD; ++j) {
    Li[j][j] = 1.0f / L[j][j];
    for (int i = j + 1; i < DD; ++i) {
      float s = 0.0f;
      for (int k = j; k < i; ++k) s += L[i][k] * Li[k][j];
      Li[i][j] = -s / L[i][i];
    }
  }

  float* dst = linv_ws + (size_t)n * DD * DD;
  for (int i = 0; i < DD; ++i)
    for (int j = 0; j < DD; ++j) dst[i * DD + j] = Li[i][j];
  logdet_ws[n] = logdet;
}

// ---------------------------------------------------------------------------
// Kernel 2: fused flow. One block per n (256 thr = 8 waves), each wave owns
// 16-point tiles. All GEMMs via v_wmma_f32_16x16x4_f32.
// ---------------------------------------------------------------------------
__global__ __launch_bounds__(256) void nf_main(
    const float* __restrict__ samples, const float* __restrict__ mu,
    const float* __restrict__ W1, const float* __restrict__ b1,
    const float* __restrict__ W2, const float* __restrict__ b2,
    const float* __restrict__ linv_ws, const float* __restrict__ logdet_ws,
    float* __restrict__ out_xf, float* __restrict__ out_ratio) {
  __shared__ __align__(16) float s_w1t[NLAYERS * HH * D2C];  // [l][j][i]
  __shared__ __align__(16) float s_w2t[NLAYERS * DD * HH];   // [l][f][k]
  __shared__ __align__(16) float s_b1[NLAYERS * HH];
  __shared__ __align__(16) float s_b2[NLAYERS * DD];
  __shared__ __align__(16) float s_linv[DD * DD];            // [i][j]
  __shared__ __align__(16) float s_mu[DD];

  const int tid = threadIdx.x;
  const int n = blockIdx.x;

  // Stage weights (transposed for A-operand reads), Linv, mu into LDS.
  for (int idx = tid; idx < NLAYERS * HH * D2C; idx += 256) {
    int l = idx >> 8, rem = idx & 255, j = rem >> 3, i = rem & 7;
    s_w1t[idx] = W1[l * (D2C * HH) + i * HH + j];
  }
  for (int idx = tid; idx < NLAYERS * DD * HH; idx += 256) {
    int l = idx >> 9, rem = idx & 511, f = rem >> 5, k = rem & 31;
    s_w2t[idx] = W2[l * (HH * DD) + k * DD + f];
  }
  if (tid < NLAYERS * HH) s_b1[tid] = b1[tid];
  if (tid < NLAYERS * DD) s_b2[tid] = b2[tid];
  if (tid < DD * DD) s_linv[tid] = linv_ws[(size_t)n * DD * DD + tid];
  if (tid < DD) s_mu[tid] = mu[n * DD + tid];
  __syncthreads();

  const float logdet_n = logdet_ws[n];

  const int lane = tid & 31;
  const int wave = tid >> 5;
  const bool hi = lane >= 16;
  const int hb = hi ? 1 : 0;
  const int col = lane & 15;

  // Tile-invariant per-lane operands.
  float mur[8];
#pragma unroll
  for (int r = 0; r < 8; ++r) mur[r] = s_mu[hb * 8 + r];
  v2f linvA[4];
#pragma unroll
  for (int c = 0; c < 4; ++c)
    linvA[c] = *(const v2f*)&s_linv[col * DD + c * 4 + hb * 2];

  for (int t = wave; t < KK / 16; t += 8) {
    const int k0 = t * 16;

    // Load raw sample tile in D layout: lane = point col, VGPR r = feature.
    const float* sp = samples + ((size_t)n * KK + k0 + col) * DD + hb * 8;
    v4f r0 = *(const v4f*)sp;
    v4f r1 = *(const v4f*)(sp + 4);
    // Prefetch this wave's next tile (t+8 -> +8*16*DD floats ahead).
    __builtin_prefetch(sp + 8 * 16 * DD, 0, 1);
    float d[8] = {r0.x, r0.y, r0.z, r0.w, r1.x, r1.y, r1.z, r1.w};

    // ---- proposal: alpha = Linv * y  (y = raw samples), chisq = |alpha|^2
    v8f acc;
#pragma unroll
    for (int r = 0; r < 8; ++r) acc[r] = 0.0f;
#pragma unroll
    for (int c = 0; c < 4; ++c) acc = wmma4(linvA[c], bchunk(d, c, hi), acc);
    float cp = 0.0f;
#pragma unroll
    for (int r = 0; r < 8; ++r) cp += acc[r] * acc[r];
    const float chisq = cp + xor16(cp);

    // ---- x = samples + mu
#pragma unroll
    for (int r = 0; r < 8; ++r) d[r] += mur[r];

    // ---- 4 coupling layers
    float ldacc = 0.0f;  // sum of s over layers (valid on lanes < 16)
#pragma unroll
    for (int l = 0; l < NLAYERS; ++l) {
      const bool flip = (l & 1) != 0;
      const int cb = flip ? 2 : 0;
      v2f ch0 = bchunk(d, cb + 0, hi);
      v2f ch1 = bchunk(d, cb + 1, hi);

      // h = tanh(x1 @ W1 + b1): 32 features = two 16-row D-layout tiles
      float h[2][8];
#pragma unroll
      for (int m = 0; m < 2; ++m) {
        v8f hc;
#pragma unroll
        for (int r = 0; r < 8; ++r) hc[r] = s_b1[l * HH + m * 16 + hb * 8 + r];
        v2f a0 = *(const v2f*)&s_w1t[l * 256 + (m * 16 + col) * 8 + 0 + hb * 2];
        v2f a1 = *(const v2f*)&s_w1t[l * 256 + (m * 16 + col) * 8 + 4 + hb * 2];
        hc = wmma4(a0, ch0, hc);
        hc = wmma4(a1, ch1, hc);
#pragma unroll
        for (int r = 0; r < 8; ++r) h[m][r] = ftanh(hc[r]);
      }

      // st = h @ W2 + b2: K=32 -> 8 accumulation chunks
      v8f st;
#pragma unroll
      for (int r = 0; r < 8; ++r) st[r] = s_b2[l * DD + hb * 8 + r];
#pragma unroll
      for (int c = 0; c < 8; ++c) {
        v2f a = *(const v2f*)&s_w2t[l * 512 + col * HH + c * 4 + hb * 2];
        st = wmma4(a, bchunk(h[c >> 2], c & 3, hi), st);
      }

      // s = tanh(st rows 0..7) lives on lanes<16; t = rows 8..15 on lanes>=16
#pragma unroll
      for (int r = 0; r < 8; ++r) {
        float tv = ftanh(st[r]);
        ldacc += tv;  // meaningful on lanes < 16 only
        float sv = flip ? tv : xor16(tv);
        float tt = flip ? xor16(st[r]) : st[r];
        float y2 = d[r] * fexp(sv) + tt;
        bool take = flip ? (!hi) : hi;
        d[r] = take ? y2 : d[r];
      }
    }

    // ---- log_target sum of squares
    float sp2 = 0.0f;
#pragma unroll
    for (int r = 0; r < 8; ++r) sp2 += d[r] * d[r];
    const float ssq = sp2 + xor16(sp2);

    // ---- store xf (16 floats per point, b128 x2 per lane)
    float* op = out_xf + ((size_t)n * KK + k0 + col) * DD + hb * 8;
    v4f o0 = {d[0], d[1], d[2], d[3]};
    v4f o1 = {d[4], d[5], d[6], d[7]};
    *(v4f*)op = o0;
    *(v4f*)(op + 4) = o1;

    // log_target - (loggauss - log_det) = 0.5*(chisq - ssq + logdet) + log_det
    if (!hi) {
      out_ratio[(size_t)n * KK + k0 + col] =
          0.5f * (chisq - ssq + logdet_n) + ldacc;
    }
  }
}

// ---------------------------------------------------------------------------
extern "C" void kernel_launch(void* const* d_in, const int* in_sizes, int n_in,
                              void* d_out, int out_size, void* d_ws,
                              size_t ws_size, hipStream_t stream) {
  (void)in_sizes; (void)n_in; (void)out_size; (void)ws_size;
  const float* samples = (const float*)d_in[0];
  const float* mu      = (const float*)d_in[1];
  const float* cov     = (const float*)d_in[2];
  const float* W1      = (const float*)d_in[3];
  const float* b1      = (const float*)d_in[4];
  const float* W2      = (const float*)d_in[5];
  const float* b2      = (const float*)d_in[6];

  float* out = (float*)d_out;
  float* out_xf = out;                              // N*K*D floats
  float* out_ratio = out + (size_t)NN * KK * DD;    // N*K floats

  float* linv_ws = (float*)d_ws;                    // N*256 floats
  float* logdet_ws = linv_ws + (size_t)NN * DD * DD;  // N floats

  nf_chol<<<NN / 256, 256, 0, stream>>>(cov, linv_ws, logdet_ws);
  nf_main<<<NN, 256, 0, stream>>>(samples, mu, W1, b1, W2, b2, linv_ws,
                                  logdet_ws, out_xf, out_ratio);
}